// pLoss_all_fidelity_50294067036168
// MI455X (gfx1250) — compile-verified
//
#include <hip/hip_runtime.h>
#include <hip/hip_bf16.h>
#include <math.h>

// D = A(16x4, f32) x B(4x16, f32) + C(16x16, f32), wave32 WMMA
typedef __attribute__((ext_vector_type(2))) float v2f;
typedef __attribute__((ext_vector_type(8))) float v8f;

#define LBL 16
#define NSTATES 1024
#define NWAVES 4
#define TPB (NWAVES * 32)
#define TILES_PER_WAVE (NSTATES / (16 * NWAVES))  // 16
#define LOG2E 1.4426950408889634f
#define SF_STRIDE 18     // padded row stride for Sf  (bank-conflict-free, keeps 8B align)
#define SFT_STRIDE 1028  // padded row stride for SfT (bank step 4 -> conflict-free)

__global__ __launch_bounds__(TPB) void crf_margin_kernel(
    const float* __restrict__ f, const float* __restrict__ y,
    const int* __restrict__ S, float* __restrict__ out,
    float* __restrict__ lossPartial)
{
  extern __shared__ float lds[];
  float* Sf  = lds;                              // [NSTATES][SF_STRIDE]  row-major S
  float* SfT = lds + NSTATES * SF_STRIDE;        // [LBL][SFT_STRIDE]     transposed S
  __shared__ float redz[NWAVES][16];             // per-wave partition partial
  __shared__ float rednum[NWAVES][16][16];       // per-wave num[label][batch]
  __shared__ float redloss[TPB];

  const int tid  = threadIdx.x;
  const int wave = tid >> 5;
  const int lane = tid & 31;
  const int hf   = lane >> 4;          // 16-lane half
  const int ln   = lane & 15;
  const int b0   = blockIdx.x * 16;    // batch base of this slab
  const int sh2  = 2 * hf;             // K-offset of this half within a K=4 chunk

  // ---- S (0/1 int32) -> f32, both row-major and transposed, in LDS ----
  for (int s = tid; s < NSTATES; s += TPB) {
    const int4* r = (const int4*)(S + (size_t)s * LBL);
    const int4 q0 = r[0], q1 = r[1], q2 = r[2], q3 = r[3];
    float v[16];
    v[0]=(float)q0.x;  v[1]=(float)q0.y;  v[2]=(float)q0.z;  v[3]=(float)q0.w;
    v[4]=(float)q1.x;  v[5]=(float)q1.y;  v[6]=(float)q1.z;  v[7]=(float)q1.w;
    v[8]=(float)q2.x;  v[9]=(float)q2.y;  v[10]=(float)q2.z; v[11]=(float)q2.w;
    v[12]=(float)q3.x; v[13]=(float)q3.y; v[14]=(float)q3.z; v[15]=(float)q3.w;
    float* row = Sf + s * SF_STRIDE;
    #pragma unroll
    for (int l = 0; l < 16; ++l) row[l] = v[l];
    #pragma unroll
    for (int l = 0; l < 16; ++l) SfT[l * SFT_STRIDE + s] = v[l];
  }

  // ---- GEMM1 B fragments (loop-invariant), pre-scaled by log2(e):
  // this half's VGPR pair for chunk j carries K = 4j+2hf, 4j+2hf+1.
  // Fixed per-batch shift M2 = log2e * sum_l max(f,0) >= any potential.
  v2f bfrag[4];
  float upart = 0.0f;
  {
    const float* fb = f + (size_t)(b0 + ln) * LBL + sh2;
    #pragma unroll
    for (int j = 0; j < 4; ++j) {
      float2 t = *(const float2*)(fb + 4 * j);
      bfrag[j].x = t.x * LOG2E;
      bfrag[j].y = t.y * LOG2E;
      upart += fmaxf(bfrag[j].x, 0.0f) + fmaxf(bfrag[j].y, 0.0f);
    }
  }
  const float M2 = upart + __shfl_xor(upart, 16, 32);  // full-row bound, both halves

  // fold the shift into the WMMA accumulator start: c = pot2 - M2 directly
  v8f cinit;
  #pragma unroll
  for (int v = 0; v < 8; ++v) cinit[v] = -M2;

  __syncthreads();

  // ---- fused double GEMM over this wave's 256 states ----
  float z_run = 0.0f;
  v8f acc2 = {0.f,0.f,0.f,0.f,0.f,0.f,0.f,0.f};  // num[label v+8*hf][batch ln]

  #pragma unroll
  for (int i = 0; i < TILES_PER_WAVE; ++i) {
    const int sb = (wave * TILES_PER_WAVE + i) * 16;  // state base of tile

    // issue A2 fragment loads early (contiguous state pairs in SfT)
    const float* acol = SfT + ln * SFT_STRIDE + sb + sh2;
    float2 a2p[4];
    #pragma unroll
    for (int j = 0; j < 4; ++j) a2p[j] = *(const float2*)(acol + 4 * j);

    // GEMM1: c = S_tile x (f^T * log2e) - M2, K=16 as 4x(K=4)
    v8f c = cinit;
    const float* arow = Sf + (size_t)(sb + ln) * SF_STRIDE + sh2;  // A row = state ln
    #pragma unroll
    for (int j = 0; j < 4; ++j) {
      float2 t = *(const float2*)(arow + 4 * j);  // ds_load_b64, conflict-free
      v2f a; a.x = t.x; a.y = t.y;
      c = __builtin_amdgcn_wmma_f32_16x16x4_f32(false, a, false, bfrag[j],
                                                (short)0, c, false, false);
    }

    // J = exp2(c); accumulate partition partial
    float jt[8];
    #pragma unroll
    for (int v = 0; v < 8; ++v) {
      jt[v] = __builtin_amdgcn_exp2f(c[v]);
      z_run += jt[v];
    }

    // each lane needs only 4 foreign J values: pre-select, then half-swap
    // half0 receives foreign jt[0],jt[1],jt[4],jt[5] (= J[8],J[9],J[12],J[13])
    // half1 receives foreign jt[2],jt[3],jt[6],jt[7] (= J[2],J[3],J[6],J[7])
    const float r0 = __shfl_xor(hf ? jt[0] : jt[2], 16, 32);
    const float r1 = __shfl_xor(hf ? jt[1] : jt[3], 16, 32);
    const float r4 = __shfl_xor(hf ? jt[4] : jt[6], 16, 32);
    const float r5 = __shfl_xor(hf ? jt[5] : jt[7], 16, 32);

    // GEMM2: num[label(M)][batch(N)] += S^T(16x4) x J(4x16)
    #pragma unroll
    for (int j = 0; j < 4; ++j) {
      v2f a2, b2;
      a2.x = a2p[j].x;   // state sb + 4j + 2hf
      a2.y = a2p[j].y;   // state sb + 4j + 2hf + 1
      if      (j == 0) { b2.x = hf ? r0    : jt[0]; b2.y = hf ? r1    : jt[1]; }
      else if (j == 1) { b2.x = hf ? r4    : jt[4]; b2.y = hf ? r5    : jt[5]; }
      else if (j == 2) { b2.x = hf ? jt[2] : r0;    b2.y = hf ? jt[3] : r1;    }
      else             { b2.x = hf ? jt[6] : r4;    b2.y = hf ? jt[7] : r5;    }
      acc2 = __builtin_amdgcn_wmma_f32_16x16x4_f32(false, a2, false, b2,
                                                   (short)0, acc2, false, false);
    }
  }

  // ---- cross-wave combine via LDS (all waves share the same shift M2) ----
  const float zc = z_run + __shfl_xor(z_run, 16, 32);
  if (hf == 0) redz[wave][ln] = zc;
  #pragma unroll
  for (int v = 0; v < 8; ++v) rednum[wave][v + 8 * hf][ln] = acc2[v];
  __syncthreads();

  // ---- finalize 256 outputs (16 labels x 16 batches) with 128 threads ----
  float lsum = 0.0f;
  #pragma unroll
  for (int r = 0; r < 2; ++r) {
    const int idx = tid + r * TPB;        // 0..255
    const int l = idx >> 4, n = idx & 15;
    float Z = 0.0f, num = 0.0f;
    #pragma unroll
    for (int w = 0; w < NWAVES; ++w) {
      Z   += redz[w][n];
      num += rednum[w][l][n];
    }
    const float p = num / Z;
    const int b = b0 + n;
    out[1 + (size_t)b * LBL + l] = p;

    const float g  = y[(size_t)b * LBL + l];
    const float fid = sqrtf(p * g + 1e-8f) + sqrtf((1.f - p) * (1.f - g) + 1e-8f);
    const float pt  = p * g + (1.f - p) * (1.f - g);
    const float at  = 0.75f * g + 0.25f * (1.f - g);
    const float omp = 1.f - pt;
    lsum += (1.f - fid) * at * omp * omp;
  }

  // ---- deterministic block loss reduction ----
  redloss[tid] = lsum;
  __syncthreads();
  #pragma unroll
  for (int s = TPB / 2; s > 0; s >>= 1) {
    if (tid < s) redloss[tid] += redloss[tid + s];
    __syncthreads();
  }
  if (tid == 0) lossPartial[blockIdx.x] = redloss[0];
}

__global__ __launch_bounds__(256) void loss_reduce_kernel(
    const float* __restrict__ partial, int nblocks, float invN,
    float* __restrict__ out)
{
  __shared__ float sh[256];
  float s = 0.0f;
  for (int i = threadIdx.x; i < nblocks; i += 256) s += partial[i];
  sh[threadIdx.x] = s;
  __syncthreads();
  #pragma unroll
  for (int k = 128; k > 0; k >>= 1) {
    if (threadIdx.x < (unsigned)k) sh[threadIdx.x] += sh[threadIdx.x + k];
    __syncthreads();
  }
  if (threadIdx.x == 0) out[0] = sh[0] * invN;
}

extern "C" void kernel_launch(void* const* d_in, const int* in_sizes, int n_in,
                              void* d_out, int out_size, void* d_ws, size_t ws_size,
                              hipStream_t stream) {
  (void)n_in; (void)out_size; (void)ws_size;
  const float* f = (const float*)d_in[0];
  const float* y = (const float*)d_in[1];
  const int*   S = (const int*)d_in[2];
  float* out = (float*)d_out;
  float* partial = (float*)d_ws;

  const int B = in_sizes[0] / LBL;      // 32768
  const int nblocks = B / 16;           // 2048
  const size_t dynLds =
      (size_t)(NSTATES * SF_STRIDE + LBL * SFT_STRIDE) * sizeof(float);  // ~136KB

  crf_margin_kernel<<<nblocks, TPB, dynLds, stream>>>(f, y, S, out, partial);
  loss_reduce_kernel<<<1, 256, 0, stream>>>(partial, nblocks,
                                            1.0f / ((float)B * (float)LBL), out);
}